// DecoderAttention_42107859370037
// MI455X (gfx1250) — compile-verified
//
#include <hip/hip_runtime.h>
#include <hip/hip_bf16.h>

typedef __attribute__((ext_vector_type(2))) float v2f;
typedef __attribute__((ext_vector_type(8))) float v8f;

#define B_  64
#define T_  50
#define S_  50
#define U_  1024
#define E_  100
#define V_  30001

// Load helpers: uniform base pointer + zero-extended 32-bit byte offset.
// This pattern maps to the GVS addressing mode (global_load ... saddr),
// avoiding per-lane 64-bit address arithmetic in the inner loop.
__device__ __forceinline__ float ldf(const char* __restrict__ base, unsigned off)
{
    return *(const float*)(base + (size_t)off);
}
__device__ __forceinline__ v2f ldv2(const char* __restrict__ base, unsigned off)
{
    return *(const v2f*)(base + (size_t)off);   // 8B-aligned at all call sites
}

// ---------------------------------------------------------------------------
// Software-pipelined WMMA f32 K-loop over an MA x NF grid of 16x16 subtiles.
// Lane-dependent addressing is folded into precomputed 32-bit BYTE offsets:
//   aoffB[mi]      = (row*lda + 2*half) * 4
//   boffB[2ni(+1)] = (2*half*ldb + col (+ldb)) * 4
// Only uniform base pointers advance per iteration (A += 16B, B += 16*ldb B).
// Next k-slice fragments are preloaded before current WMMAs issue.
// K must be a positive multiple of 4 (holds for all call sites: 100/1024).
// ---------------------------------------------------------------------------
template <int MA, int NF>
__device__ __forceinline__ void gemm_tile(
    const float* __restrict__ A, const unsigned* __restrict__ aoffB, int K,
    const float* __restrict__ Bm, int ldb, const unsigned* __restrict__ boffB,
    v8f* __restrict__ acc)
{
    const char* __restrict__ Ak = (const char*)A;
    const char* __restrict__ Bk = (const char*)Bm;
    const size_t bstep = (size_t)16 * (size_t)ldb;   // bytes per 4 K-rows

    v2f a[MA], b[NF];
    #pragma unroll
    for (int mi = 0; mi < MA; ++mi) a[mi] = ldv2(Ak, aoffB[mi]);
    #pragma unroll
    for (int ni = 0; ni < NF; ++ni) {
        b[ni].x = ldf(Bk, boffB[2 * ni]);
        b[ni].y = ldf(Bk, boffB[2 * ni + 1]);
    }

    #pragma unroll 2
    for (int k0 = 0; k0 + 4 < K; k0 += 4) {
        const char* __restrict__ An = Ak + 16;
        const char* __restrict__ Bn = Bk + bstep;
        v2f an[MA], bn[NF];
        #pragma unroll
        for (int mi = 0; mi < MA; ++mi) an[mi] = ldv2(An, aoffB[mi]);
        #pragma unroll
        for (int ni = 0; ni < NF; ++ni) {
            bn[ni].x = ldf(Bn, boffB[2 * ni]);
            bn[ni].y = ldf(Bn, boffB[2 * ni + 1]);
        }

        #pragma unroll
        for (int mi = 0; mi < MA; ++mi)
            #pragma unroll
            for (int ni = 0; ni < NF; ++ni)
                acc[mi * NF + ni] = __builtin_amdgcn_wmma_f32_16x16x4_f32(
                    false, a[mi], false, b[ni], (short)0, acc[mi * NF + ni], false, false);

        #pragma unroll
        for (int mi = 0; mi < MA; ++mi) a[mi] = an[mi];
        #pragma unroll
        for (int ni = 0; ni < NF; ++ni) b[ni] = bn[ni];
        Ak = An; Bk = Bn;
    }

    #pragma unroll
    for (int mi = 0; mi < MA; ++mi)
        #pragma unroll
        for (int ni = 0; ni < NF; ++ni)
            acc[mi * NF + ni] = __builtin_amdgcn_wmma_f32_16x16x4_f32(
                false, a[mi], false, b[ni], (short)0, acc[mi * NF + ni], false, false);
}

// ---------------------------------------------------------------------------
// Generic f32 WMMA GEMM:  D[M,N] = A[M,K] @ B[K,N] (+ bias[n])
// Block 256 threads = 8 waves (2 along M x 4 along N); wave tile 32x64;
// block tile 64x256.  Edge columns clamped at load, suppressed at store.
// ---------------------------------------------------------------------------
__global__ __launch_bounds__(256) void wmma_gemm_f32(
    const float* __restrict__ A, int lda,
    const float* __restrict__ Bm, int ldb,
    float* __restrict__ D, int ldd,
    int M, int N, int K,
    const float* __restrict__ bias)
{
    const int lane = threadIdx.x & 31;
    const int wave = threadIdx.x >> 5;
    const int half = lane >> 4;
    const int r    = lane & 15;

    const int mbase = blockIdx.y * 64  + (wave & 1) * 32;
    const int nbase = blockIdx.x * 256 + (wave >> 1) * 64;
    if (mbase >= M || nbase >= N) return;          // wave-uniform exit

    v8f acc[8];
    #pragma unroll
    for (int i = 0; i < 8; ++i) acc[i] = (v8f){0,0,0,0,0,0,0,0};

    unsigned aoffB[2];
    #pragma unroll
    for (int mi = 0; mi < 2; ++mi) {
        int row = mbase + mi * 16 + r;
        if (row > M - 1) row = M - 1;              // M always 16-aligned here
        aoffB[mi] = (unsigned)(row * lda + 2 * half) * 4u;
    }

    int rcols[4];
    unsigned boffB[8];
    #pragma unroll
    for (int ni = 0; ni < 4; ++ni) {
        rcols[ni] = nbase + 16 * ni + r;
        const int cc = rcols[ni] < N ? rcols[ni] : (N - 1);   // branchless clamp
        boffB[2 * ni]     = (unsigned)(2 * half * ldb + cc) * 4u;
        boffB[2 * ni + 1] = boffB[2 * ni] + (unsigned)ldb * 4u;
    }

    __builtin_prefetch((const char*)Bm + boffB[0], 0, 0);
    gemm_tile<2, 4>(A, aoffB, K, Bm, ldb, boffB, acc);

    #pragma unroll
    for (int ni = 0; ni < 4; ++ni) {
        if (rcols[ni] >= N) continue;              // store guard only
        const int c = rcols[ni];
        const float bv = bias ? bias[c] : 0.0f;
        #pragma unroll
        for (int mi = 0; mi < 2; ++mi)
            #pragma unroll
            for (int rv = 0; rv < 8; ++rv) {
                const int row = mbase + mi * 16 + rv + 8 * half;
                D[(size_t)row * (size_t)ldd + c] = acc[mi * 4 + ni][rv] + bv;
            }
    }
}

// ---------------------------------------------------------------------------
// Fused recurrent step:  z = x_t @ WxE + attn @ WxA + h @ Uh + b, then LSTM
// gates in-register.  Wave tile: 16 rows x 32 cols in EACH of the 4 gate
// regions (i,f,g,o), so gate quadruples land in matching acc slots.
// Grid: 16 blocks (64 u-cols each), 8 waves (4 along M=64, 2 along u).
// ---------------------------------------------------------------------------
__global__ __launch_bounds__(256) void lstm_step_fused(
    const float* __restrict__ xemb_t, int ldx,
    const float* __restrict__ attn,
    const float* __restrict__ hin,
    const float* __restrict__ WxE,
    const float* __restrict__ WxA,
    const float* __restrict__ Uh,
    const float* __restrict__ bvec,
    float* __restrict__ c,
    float* __restrict__ hout)
{
    const int lane = threadIdx.x & 31;
    const int wave = threadIdx.x >> 5;
    const int half = lane >> 4;
    const int r    = lane & 15;

    const int m0 = (wave & 3) * 16;                       // 0..48 (M = 64)
    const int u0 = blockIdx.x * 64 + (wave >> 2) * 32;    // 0..992

    v8f acc[8];                                           // acc[g*2+j]
    #pragma unroll
    for (int i = 0; i < 8; ++i) acc[i] = (v8f){0,0,0,0,0,0,0,0};

    unsigned boffB[16];
    #pragma unroll
    for (int g = 0; g < 4; ++g)
        #pragma unroll
        for (int j = 0; j < 2; ++j) {
            const int col = g * U_ + u0 + 16 * j + r;     // in-range (4U = 4096)
            boffB[2 * (g * 2 + j)]     = (unsigned)(2 * half * (4 * U_) + col) * 4u;
            boffB[2 * (g * 2 + j) + 1] = boffB[2 * (g * 2 + j)] + 4u * U_ * 4u;
        }

    const int arow = m0 + r;                              // < 64 always
    const unsigned aoffX = (unsigned)(arow * ldx + 2 * half) * 4u;
    const unsigned aoffU = (unsigned)(arow * U_  + 2 * half) * 4u;

    gemm_tile<1, 8>(xemb_t, &aoffX, E_, WxE, 4 * U_, boffB, acc);
    gemm_tile<1, 8>(attn,   &aoffU, U_, WxA, 4 * U_, boffB, acc);
    gemm_tile<1, 8>(hin,    &aoffU, U_, Uh,  4 * U_, boffB, acc);

    #pragma unroll
    for (int j = 0; j < 2; ++j) {
        const int u = u0 + 16 * j + r;
        const float bi = bvec[u];
        const float bf = bvec[U_ + u];
        const float bg = bvec[2 * U_ + u];
        const float bo = bvec[3 * U_ + u];
        #pragma unroll
        for (int rv = 0; rv < 8; ++rv) {
            const int row = m0 + rv + 8 * half;
            const size_t idx = (size_t)row * U_ + u;
            const float zi = acc[0 * 2 + j][rv] + bi;
            const float zf = acc[1 * 2 + j][rv] + bf;
            const float zg = acc[2 * 2 + j][rv] + bg;
            const float zo = acc[3 * 2 + j][rv] + bo;
            const float si = 1.0f / (1.0f + expf(-zi));
            const float sf = 1.0f / (1.0f + expf(-zf));
            const float so = 1.0f / (1.0f + expf(-zo));
            const float cn = sf * c[idx] + si * tanhf(zg);
            c[idx]    = cn;
            hout[idx] = so * tanhf(cn);
        }
    }
}

// ---------------------------------------------------------------------------
// Fused attention output:  attn_new = h @ Wa[0:U] + ctx @ Wa[U:2U],
// mirrored into attn_all[b*T + t].  Wave tile 32x64; block tile 64x256.
// Grid: 4 blocks (N = 1024).
// ---------------------------------------------------------------------------
__global__ __launch_bounds__(256) void attn_out_fused(
    const float* __restrict__ h,
    const float* __restrict__ ctx,
    const float* __restrict__ WaH,
    const float* __restrict__ WaC,
    float* __restrict__ attn,
    float* __restrict__ attn_all_t, int ld2)
{
    const int lane = threadIdx.x & 31;
    const int wave = threadIdx.x >> 5;
    const int half = lane >> 4;
    const int r    = lane & 15;

    const int mbase = (wave & 1) * 32;                      // M = 64
    const int nbase = blockIdx.x * 256 + (wave >> 1) * 64;  // N = 1024

    v8f acc[8];
    #pragma unroll
    for (int i = 0; i < 8; ++i) acc[i] = (v8f){0,0,0,0,0,0,0,0};

    unsigned aoffB[2];
    #pragma unroll
    for (int mi = 0; mi < 2; ++mi)
        aoffB[mi] = (unsigned)((mbase + mi * 16 + r) * U_ + 2 * half) * 4u;

    int cols[4];
    unsigned boffB[8];
    #pragma unroll
    for (int ni = 0; ni < 4; ++ni) {
        cols[ni] = nbase + 16 * ni + r;                     // in-range
        boffB[2 * ni]     = (unsigned)(2 * half * U_ + cols[ni]) * 4u;
        boffB[2 * ni + 1] = boffB[2 * ni] + (unsigned)U_ * 4u;
    }

    gemm_tile<2, 4>(h,   aoffB, U_, WaH, U_, boffB, acc);
    gemm_tile<2, 4>(ctx, aoffB, U_, WaC, U_, boffB, acc);

    #pragma unroll
    for (int ni = 0; ni < 4; ++ni) {
        const int cidx = cols[ni];
        #pragma unroll
        for (int mi = 0; mi < 2; ++mi)
            #pragma unroll
            for (int rv = 0; rv < 8; ++rv) {
                const int row = mbase + mi * 16 + rv + 8 * half;
                const float v = acc[mi * 4 + ni][rv];
                attn[(size_t)row * U_ + cidx] = v;
                attn_all_t[(size_t)row * ld2 + cidx] = v;
            }
    }
}

// ---------------------------------------------------------------------------
__global__ void gather_emb(const int* __restrict__ tok, const float* __restrict__ emb,
                           float* __restrict__ xemb, int n)
{
    int i = blockIdx.x * blockDim.x + threadIdx.x;
    if (i >= n) return;
    int rt = i / E_;
    int e  = i - rt * E_;
    xemb[i] = emb[(size_t)tok[rt] * E_ + e];
}

__global__ void init_state(const float* __restrict__ h0, const float* __restrict__ c0,
                           float* __restrict__ h, float* __restrict__ c, float* __restrict__ attn)
{
    int i = blockIdx.x * blockDim.x + threadIdx.x;
    if (i >= B_ * U_) return;
    h[i] = h0[i];
    c[i] = c0[i];
    attn[i] = 0.0f;
}

// One block per batch row: score = h . keys[b,s,:], softmax over S, ctx = align . memory
__global__ __launch_bounds__(256) void attention_step(
    const float* __restrict__ h, const float* __restrict__ keys,
    const float* __restrict__ memory, float* __restrict__ ctx)
{
    __shared__ float sAlign[S_];
    const int b    = blockIdx.x;
    const int tid  = threadIdx.x;
    const int lane = tid & 31;
    const int wave = tid >> 5;
    const float* __restrict__ hb = h + (size_t)b * U_;

    for (int s = wave; s < S_; s += 8) {
        const float* __restrict__ kr = keys + ((size_t)b * S_ + s) * U_;
        float acc = 0.0f;
        for (int u = lane; u < U_; u += 32) acc += hb[u] * kr[u];
        for (int off = 16; off; off >>= 1) acc += __shfl_xor(acc, off, 32);
        if (lane == 0) sAlign[s] = acc;
    }
    __syncthreads();
    if (tid == 0) {
        float mx = -INFINITY;
        for (int s = 0; s < S_; ++s) mx = fmaxf(mx, sAlign[s]);
        float sum = 0.0f;
        for (int s = 0; s < S_; ++s) { float e = expf(sAlign[s] - mx); sAlign[s] = e; sum += e; }
        float inv = 1.0f / sum;
        for (int s = 0; s < S_; ++s) sAlign[s] *= inv;
    }
    __syncthreads();
    for (int u = tid; u < U_; u += 256) {
        float acc = 0.0f;
        const float* __restrict__ mb = memory + (size_t)b * S_ * U_ + u;
        #pragma unroll 5
        for (int s = 0; s < S_; ++s) acc += sAlign[s] * mb[(size_t)s * U_];
        ctx[(size_t)b * U_ + u] = acc;
    }
}

// ---------------------------------------------------------------------------
extern "C" void kernel_launch(void* const* d_in, const int* in_sizes, int n_in,
                              void* d_out, int out_size, void* d_ws, size_t ws_size,
                              hipStream_t stream)
{
    const int*   tokens = (const int*)  d_in[0];
    const float* memory = (const float*)d_in[1];
    const float* h0     = (const float*)d_in[2];
    const float* c0     = (const float*)d_in[3];
    const float* emb    = (const float*)d_in[4];
    const float* Wx     = (const float*)d_in[5];
    const float* Uh     = (const float*)d_in[6];
    const float* bvec   = (const float*)d_in[7];
    const float* Wm     = (const float*)d_in[8];
    const float* Wa     = (const float*)d_in[9];
    const float* Wo     = (const float*)d_in[10];
    const float* bo     = (const float*)d_in[11];
    float*       out    = (float*)d_out;

    float* ws = (float*)d_ws;
    float* keys     = ws; ws += (size_t)B_ * S_ * U_;
    float* xemb     = ws; ws += (size_t)B_ * T_ * E_;
    float* hA       = ws; ws += (size_t)B_ * U_;
    float* hB       = ws; ws += (size_t)B_ * U_;
    float* cbuf     = ws; ws += (size_t)B_ * U_;
    float* attn     = ws; ws += (size_t)B_ * U_;
    float* ctx      = ws; ws += (size_t)B_ * U_;
    float* attn_all = ws; ws += (size_t)B_ * T_ * U_;

    const float* WxE = Wx;                          // rows [0,E)
    const float* WxA = Wx + (size_t)E_ * 4 * U_;    // rows [E,E+U)
    const float* WaH = Wa;                          // rows [0,U)
    const float* WaC = Wa + (size_t)U_ * U_;        // rows [U,2U)

    dim3 blk(256);

    // keys = memory @ Wm   [3200x1024] x [1024x1024]
    {
        dim3 g((U_ + 255) / 256, (B_ * S_ + 63) / 64);
        wmma_gemm_f32<<<g, blk, 0, stream>>>(memory, U_, Wm, U_, keys, U_,
                                             B_ * S_, U_, U_, nullptr);
    }
    gather_emb<<<(B_ * T_ * E_ + 255) / 256, blk, 0, stream>>>(tokens, emb, xemb, B_ * T_ * E_);
    init_state<<<(B_ * U_ + 255) / 256, blk, 0, stream>>>(h0, c0, hA, cbuf, attn);

    float* hin  = hA;
    float* hout = hB;
    for (int t = 0; t < T_; ++t) {
        lstm_step_fused<<<16, blk, 0, stream>>>(xemb + (size_t)t * E_, T_ * E_,
                                                attn, hin, WxE, WxA, Uh, bvec,
                                                cbuf, hout);
        attention_step<<<B_, blk, 0, stream>>>(hout, keys, memory, ctx);
        attn_out_fused<<<4, blk, 0, stream>>>(hout, ctx, WaH, WaC, attn,
                                              attn_all + (size_t)t * U_, T_ * U_);
        float* tmp = hin; hin = hout; hout = tmp;   // ping-pong h
    }

    // logits = attn_all @ Wo + bo   [3200x1024] x [1024x30001]
    {
        dim3 g((V_ + 255) / 256, (B_ * T_ + 63) / 64);
        wmma_gemm_f32<<<g, blk, 0, stream>>>(attn_all, U_, Wo, V_, out, V_,
                                             B_ * T_, V_, U_, bo);
    }
}